// ResidualAttentionWithSyntacticDistanceBlock_45775761440793
// MI455X (gfx1250) — compile-verified
//
#include <hip/hip_runtime.h>
#include <hip/hip_bf16.h>
#include <math.h>
#include <stdint.h>

typedef __attribute__((ext_vector_type(2))) float v2f;
typedef __attribute__((ext_vector_type(8))) float v8f;

#define B_ 4
#define S_ 1024
#define E_ 512
#define H_ 8

#if __has_builtin(__builtin_amdgcn_tensor_load_to_lds) && __has_builtin(__builtin_amdgcn_s_wait_tensorcnt)
#define HAVE_TDM 1
#else
#define HAVE_TDM 0
#endif

#if HAVE_TDM
typedef __attribute__((ext_vector_type(4))) unsigned int tdm_u4;
typedef __attribute__((ext_vector_type(8))) int tdm_i8;
typedef __attribute__((ext_vector_type(4))) int tdm_i4;

// 2-D tile load via Tensor Data Mover: tile_x contiguous elems per row, tile_y rows,
// row stride = stride_elems (f32), LDS rows padded by pad_amount DWORDs every
// (2<<pad_interval_code) DWORDs. Issued once per calling wave; tracked by TENSORcnt.
__device__ __forceinline__ void tdm_load_2d(unsigned lds_byte_addr, const float* gaddr,
                                            int tile_x, int tile_y, long stride_elems,
                                            int pad_interval_code, int pad_amount_code) {
    unsigned long long ga = (unsigned long long)(uintptr_t)gaddr;
    tdm_u4 g0;
    g0[0] = 1u;                                            // count=1 (valid user D#)
    g0[1] = lds_byte_addr;                                 // lds_addr
    g0[2] = (unsigned)(ga & 0xFFFFFFFFu);                  // global_addr[31:0]
    g0[3] = (unsigned)((ga >> 32) & 0x1FFFFFFu) | (2u << 30);  // addr[56:32] | type=2
    tdm_i8 g1;
    g1[0] = (2 << 16) | (1 << 20) | (pad_interval_code << 22) | (pad_amount_code << 25);
    g1[1] = (tile_x & 0xFFFF) << 16;                       // tensor_dim0[15:0]
    g1[2] = ((tile_x >> 16) & 0xFFFF) | ((tile_y & 0xFFFF) << 16);  // dim0 hi | dim1 lo
    g1[3] = ((tile_y >> 16) & 0xFFFF) | ((tile_x & 0xFFFF) << 16);  // dim1 hi | tile_dim0
    g1[4] = (tile_y & 0xFFFF);                             // tile_dim1 (tile_dim2=0)
    g1[5] = (int)(stride_elems & 0xFFFFFFFFL);             // tensor_dim0_stride[31:0]
    g1[6] = (int)((stride_elems >> 32) & 0xFFFFL);         // stride[47:32] | dim1_stride lo=0
    g1[7] = 0;
    tdm_i4 z4 = {0, 0, 0, 0};
#if __clang_major__ >= 23
    tdm_i8 z8 = {0, 0, 0, 0, 0, 0, 0, 0};
    __builtin_amdgcn_tensor_load_to_lds(g0, g1, z4, z4, z8, 0);
#else
    __builtin_amdgcn_tensor_load_to_lds(g0, g1, z4, z4, 0);
#endif
}
#endif  // HAVE_TDM

__device__ __forceinline__ float wave_sum(float v) {
    for (int off = 16; off > 0; off >>= 1) v += __shfl_xor(v, off, 32);
    return v;
}

// ---------------- LayerNorm: one wave32 per row of 512 ----------------
__global__ __launch_bounds__(256) void ln_kernel(const float* __restrict__ x,
                                                 const float* __restrict__ g,
                                                 const float* __restrict__ b,
                                                 float* __restrict__ out, int rows) {
    int wave = (blockIdx.x * blockDim.x + threadIdx.x) >> 5;
    int lane = threadIdx.x & 31;
    if (wave >= rows) return;
    const float* xr = x + (size_t)wave * E_;
    float vals[16];
    float s = 0.f;
#pragma unroll
    for (int i = 0; i < 16; i++) { vals[i] = xr[lane + 32 * i]; s += vals[i]; }
    s = wave_sum(s);
    float mean = s * (1.0f / E_);
    float vs = 0.f;
#pragma unroll
    for (int i = 0; i < 16; i++) { float t = vals[i] - mean; vs += t * t; }
    vs = wave_sum(vs);
    float inv = rsqrtf(vs * (1.0f / E_) + 1e-5f);
    float* orow = out + (size_t)wave * E_;
#pragma unroll
    for (int i = 0; i < 16; i++) {
        int c = lane + 32 * i;
        orow[c] = (vals[i] - mean) * inv * g[c] + b[c];
    }
}

// ------------- syntactic distance: d[b,h,s] = tanh(conv) (wave per output) -------------
__global__ __launch_bounds__(256) void dist_kernel(const float* __restrict__ nq,
                                                   const float* __restrict__ conv_w,
                                                   const float* __restrict__ conv_b,
                                                   float* __restrict__ d) {
    int gw = (blockIdx.x * blockDim.x + threadIdx.x) >> 5;  // (b*H + h)*S + s
    int lane = threadIdx.x & 31;
    if (gw >= B_ * H_ * S_) return;
    int s = gw & (S_ - 1);
    int h = (gw >> 10) & (H_ - 1);
    int b = gw >> 13;
    float acc = 0.f;
#pragma unroll
    for (int k = 0; k < 3; k++) {
        int src = s + k - 2;
        if (src < 0) continue;
        const float* xr = nq + ((size_t)b * S_ + src) * E_;
        for (int e = lane; e < E_; e += 32)
            acc += xr[e] * conv_w[(h * E_ + e) * 3 + k];
    }
    acc = wave_sum(acc);
    if (lane == 0) d[gw] = tanhf(acc + conv_b[h]);
}

// ---------------- generic fp32 WMMA GEMM: C = A @ B^T (+bias)(+add), epilogues ----------------
// A: [M,K] row-major (lda). Bm: [N,K] row-major (ldb). Tile 128x64, 8 waves, K-chunk 16.
// TDM path: double-buffered LDS, wave 0 drives tensor_load_to_lds, one barrier/iteration.
__global__ __launch_bounds__(256) void gemm_f32_wmma(
    const float* __restrict__ A, long batchA,
    const float* __restrict__ Bm, long batchB,
    const float* __restrict__ bias,
    const float* __restrict__ add,
    float* __restrict__ C, long batchC,
    int N, int Kd, int lda, int ldb, int ldc,
    int mode, float scale, int scale_lt) {
    int bz = blockIdx.z;
    A += (size_t)bz * batchA;
    Bm += (size_t)bz * batchB;
    C += (size_t)bz * batchC;
    int m0 = blockIdx.y * 128, n0 = blockIdx.x * 64;
    int t = threadIdx.x, wave = t >> 5, lane = t & 31;

    v8f acc[4];
#pragma unroll
    for (int nt = 0; nt < 4; nt++)
#pragma unroll
        for (int r = 0; r < 8; r++) acc[nt][r] = 0.f;

    int arow = wave * 16 + (lane & 15);  // local A row for WMMA
    int koff = (lane >> 4) * 2;          // K-pair offset from lane half
    int nlane = lane & 15;

#if HAVE_TDM
    __shared__ float Alds[2][128][20];
    __shared__ float Blds[2][64][20];
    int niter = Kd >> 4;
    if (wave == 0) {  // wave-uniform branch: TDM ignores EXEC, must be scalar-skipped
        tdm_load_2d((unsigned)(uintptr_t)&Alds[0][0][0], A + (size_t)m0 * lda, 16, 128, lda, 3, 3);
        tdm_load_2d((unsigned)(uintptr_t)&Blds[0][0][0], Bm + (size_t)n0 * ldb, 16, 64, ldb, 3, 3);
        __builtin_amdgcn_s_wait_tensorcnt(0);
    }
    __syncthreads();
    for (int it = 0; it < niter; it++) {
        int cur = it & 1;
        if (wave == 0 && it + 1 < niter) {  // prefetch next K-chunk into other buffer
            int k1 = (it + 1) << 4;
            tdm_load_2d((unsigned)(uintptr_t)&Alds[cur ^ 1][0][0], A + (size_t)m0 * lda + k1, 16, 128, lda, 3, 3);
            tdm_load_2d((unsigned)(uintptr_t)&Blds[cur ^ 1][0][0], Bm + (size_t)n0 * ldb + k1, 16, 64, ldb, 3, 3);
        }
#pragma unroll
        for (int kk = 0; kk < 16; kk += 4) {
            v2f a;
            a.x = Alds[cur][arow][kk + koff];
            a.y = Alds[cur][arow][kk + koff + 1];
#pragma unroll
            for (int nt = 0; nt < 4; nt++) {
                v2f bv;
                bv.x = Blds[cur][nt * 16 + nlane][kk + koff];
                bv.y = Blds[cur][nt * 16 + nlane][kk + koff + 1];
                acc[nt] = __builtin_amdgcn_wmma_f32_16x16x4_f32(
                    false, a, false, bv, (short)0, acc[nt], false, false);
            }
        }
        if (wave == 0 && it + 1 < niter) __builtin_amdgcn_s_wait_tensorcnt(0);
        __syncthreads();
    }
#else
    __shared__ float Alds[128][20];
    __shared__ float Blds[64][20];
    int ar = t >> 1, acb = (t & 1) * 8;
    int br = t >> 2, bcb = (t & 3) * 4;
    for (int k0 = 0; k0 < Kd; k0 += 16) {
        {
            const float* src = A + (size_t)(m0 + ar) * lda + k0 + acb;
            float4 p0 = *(const float4*)(src);
            float4 p1 = *(const float4*)(src + 4);
            *(float4*)&Alds[ar][acb] = p0;
            *(float4*)&Alds[ar][acb + 4] = p1;
        }
        {
            const float* src = Bm + (size_t)(n0 + br) * ldb + k0 + bcb;
            float4 p = *(const float4*)(src);
            *(float4*)&Blds[br][bcb] = p;
        }
        if (k0 + 16 < Kd) {
            __builtin_prefetch(A + (size_t)(m0 + ar) * lda + (k0 + 16) + acb, 0, 0);
            __builtin_prefetch(Bm + (size_t)(n0 + br) * ldb + (k0 + 16) + bcb, 0, 0);
        }
        __syncthreads();
#pragma unroll
        for (int kk = 0; kk < 16; kk += 4) {
            v2f a;
            a.x = Alds[arow][kk + koff];
            a.y = Alds[arow][kk + koff + 1];
#pragma unroll
            for (int nt = 0; nt < 4; nt++) {
                v2f bv;
                bv.x = Blds[nt * 16 + nlane][kk + koff];
                bv.y = Blds[nt * 16 + nlane][kk + koff + 1];
                acc[nt] = __builtin_amdgcn_wmma_f32_16x16x4_f32(
                    false, a, false, bv, (short)0, acc[nt], false, false);
            }
        }
        __syncthreads();
    }
#endif

    int mbase = m0 + wave * 16 + ((lane >> 4) * 8);
#pragma unroll
    for (int nt = 0; nt < 4; nt++) {
        int n = n0 + nt * 16 + nlane;
        float bv = bias ? bias[n] : 0.f;
#pragma unroll
        for (int r = 0; r < 8; r++) {
            int m = mbase + r;
            float v = acc[nt][r] + bv;
            if (mode == 1) v = 0.5f * v * (1.f + erff(v * 0.70710678118654752f));
            else if (mode == 2 && n < scale_lt) v *= scale;
            if (add) v += add[(size_t)m * ldc + n];
            C[(size_t)m * ldc + n] = v;
        }
    }
}

// ---------------- gated attention: ctx[b,i,h*E+e] = (P @ V)/den, P built on the fly ----------------
// grid: (E/128, S/32, B*H). 256 threads = 8 waves; wave = its 16-col e-subtile.
// V tile staged by TDM (overlapped with sigmoid-gated P construction) when available.
__global__ __launch_bounds__(256) void attn_kernel(const float* __restrict__ scores,
                                                   const float* __restrict__ qkv,
                                                   const float* __restrict__ d,
                                                   float* __restrict__ ctx) {
    __shared__ float P[32][36];
    __shared__ float V[32][132];
    __shared__ float den[32];
    __shared__ float di[32];
    int t = threadIdx.x, wave = t >> 5, lane = t & 31;
    int e0 = blockIdx.x * 128, i0 = blockIdx.y * 32;
    int bh = blockIdx.z;
    int b = bh >> 3, h = bh & 7;
    const float* sc = scores + ((size_t)b * S_ + i0) * S_;
    const float* dptr = d + (size_t)bh * S_;
    const float* vmat = qkv + (size_t)b * S_ * (3 * E_) + 2 * E_;  // v = cols [1024,1536)

    if (t < 32) { den[t] = 0.f; di[t] = dptr[i0 + t]; }
    __syncthreads();

    v8f acc[2];
#pragma unroll
    for (int m = 0; m < 2; m++)
#pragma unroll
        for (int r = 0; r < 8; r++) acc[m][r] = 0.f;

    int pr = t >> 3, pc4 = (t & 7) * 4;  // P build coords
    int koff = (lane >> 4) * 2;
    int rowA = lane & 15, colB = lane & 15;
#if !HAVE_TDM
    int vr = t >> 3, vseg = t & 7;       // V stage coords (manual path)
#endif

    for (int j0 = 0; j0 < S_; j0 += 32) {
#if HAVE_TDM
        if (wave == 0)  // async V[32][128] tile (row stride 1536 f32, LDS rows padded to 132)
            tdm_load_2d((unsigned)(uintptr_t)&V[0][0],
                        vmat + (size_t)j0 * (3 * E_) + e0, 128, 32, 3 * E_, 6, 3);
#else
        {
            const float* src = vmat + (size_t)(j0 + vr) * (3 * E_) + e0 + vseg * 16;
            float4 x0 = *(const float4*)(src + 0);
            float4 x1 = *(const float4*)(src + 4);
            float4 x2 = *(const float4*)(src + 8);
            float4 x3 = *(const float4*)(src + 12);
            float4* dst = (float4*)&V[vr][vseg * 16];
            dst[0] = x0; dst[1] = x1; dst[2] = x2; dst[3] = x3;
        }
#endif
        {   // build P = scores * sigmoid(d_i - d_j); accumulate row denominators
            float dij = di[pr];
            float psum = 0.f;
#pragma unroll
            for (int u = 0; u < 4; u++) {
                int c = pc4 + u;
                float g = 1.f / (1.f + __expf(-(dij - dptr[j0 + c])));
                float p = sc[(size_t)pr * S_ + j0 + c] * g;
                P[pr][c] = p;
                psum += p;
            }
            atomicAdd(&den[pr], psum);
        }
#if HAVE_TDM
        if (wave == 0) __builtin_amdgcn_s_wait_tensorcnt(0);
#endif
        __syncthreads();
#pragma unroll
        for (int kk = 0; kk < 32; kk += 4) {
            v2f bv;
            bv.x = V[kk + koff][wave * 16 + colB];
            bv.y = V[kk + koff + 1][wave * 16 + colB];
#pragma unroll
            for (int ms = 0; ms < 2; ms++) {
                v2f a;
                a.x = P[ms * 16 + rowA][kk + koff];
                a.y = P[ms * 16 + rowA][kk + koff + 1];
                acc[ms] = __builtin_amdgcn_wmma_f32_16x16x4_f32(
                    false, a, false, bv, (short)0, acc[ms], false, false);
            }
        }
        __syncthreads();
    }

    int mbase = (lane >> 4) * 8;
#pragma unroll
    for (int ms = 0; ms < 2; ms++) {
#pragma unroll
        for (int r = 0; r < 8; r++) {
            int li = ms * 16 + mbase + r;
            float dd = den[li] + 1e-12f;
            int i = i0 + li;
            int col = h * E_ + e0 + wave * 16 + (lane & 15);
            ctx[((size_t)(b * S_ + i)) * (H_ * E_) + col] = acc[ms][r] / dd;
        }
    }
}

extern "C" void kernel_launch(void* const* d_in, const int* in_sizes, int n_in,
                              void* d_out, int out_size, void* d_ws, size_t ws_size,
                              hipStream_t stream) {
    const float* query     = (const float*)d_in[0];
    const float* ln1_g     = (const float*)d_in[1];
    const float* ln1_b     = (const float*)d_in[2];
    const float* in_proj_w = (const float*)d_in[3];
    const float* in_proj_b = (const float*)d_in[4];
    const float* out_w     = (const float*)d_in[5];
    const float* out_b     = (const float*)d_in[6];
    const float* conv_w    = (const float*)d_in[7];
    const float* conv_b    = (const float*)d_in[8];
    const float* ln2_g     = (const float*)d_in[9];
    const float* ln2_b     = (const float*)d_in[10];
    const float* mlp_w1    = (const float*)d_in[11];
    const float* mlp_b1    = (const float*)d_in[12];
    const float* mlp_w2    = (const float*)d_in[13];
    const float* mlp_b2    = (const float*)d_in[14];
    float* out = (float*)d_out;

    float* ws     = (float*)d_ws;
    float* nq     = ws;                  // 4096*512
    float* dist   = ws + 2097152;        // 32768
    float* qkv    = ws + 2129920;        // 4096*1536
    float* scores = ws + 8421376;        // 4*1024*1024
    float* ctx    = ws + 12615680;       // 4096*4096
    float* x1     = ws + 29392896;       // 4096*512
    float* hin    = ws + 31490048;       // 4096*512
    float* hmid   = ws;                  // 4096*2048 (reuses dead nq/qkv region)

    const int rows = B_ * S_;            // 4096
    const float qscale = 0.04419417382415922f;  // 1/sqrt(512)

    // 1) LN1
    ln_kernel<<<dim3(512), dim3(256), 0, stream>>>(query, ln1_g, ln1_b, nq, rows);
    // 2) syntactic distances
    dist_kernel<<<dim3(4096), dim3(256), 0, stream>>>(nq, conv_w, conv_b, dist);
    // 3) qkv = nq @ W^T + b ; q pre-scaled by 1/sqrt(E)
    gemm_f32_wmma<<<dim3(24, 32, 1), dim3(256), 0, stream>>>(
        nq, 0, in_proj_w, 0, in_proj_b, nullptr, qkv, 0,
        1536, 512, 512, 512, 1536, 2, qscale, 512);
    // 4) scores[b] = q[b] @ k[b]^T  (batched over B)
    gemm_f32_wmma<<<dim3(16, 8, 4), dim3(256), 0, stream>>>(
        qkv, (long)S_ * 1536, qkv + 512, (long)S_ * 1536, nullptr, nullptr,
        scores, (long)S_ * S_,
        1024, 512, 1536, 1536, 1024, 0, 0.f, 0);
    // 5) gated attention -> ctx in [B,S,H*E] layout
    attn_kernel<<<dim3(4, 32, 32), dim3(256), 0, stream>>>(scores, qkv, dist, ctx);
    // 6) x1 = query + ctx @ out_w^T + out_b
    gemm_f32_wmma<<<dim3(8, 32, 1), dim3(256), 0, stream>>>(
        ctx, 0, out_w, 0, out_b, query, x1, 0,
        512, 4096, 4096, 4096, 512, 0, 0.f, 0);
    // 7) LN2
    ln_kernel<<<dim3(512), dim3(256), 0, stream>>>(x1, ln2_g, ln2_b, hin, rows);
    // 8) hmid = gelu(hin @ w1^T + b1)
    gemm_f32_wmma<<<dim3(32, 32, 1), dim3(256), 0, stream>>>(
        hin, 0, mlp_w1, 0, mlp_b1, nullptr, hmid, 0,
        2048, 512, 512, 512, 2048, 1, 0.f, 0);
    // 9) out = x1 + hmid @ w2^T + b2
    gemm_f32_wmma<<<dim3(8, 32, 1), dim3(256), 0, stream>>>(
        hmid, 0, mlp_w2, 0, mlp_b2, x1, out, 0,
        512, 2048, 2048, 2048, 512, 0, 0.f, 0);
    (void)in_sizes; (void)n_in; (void)out_size; (void)ws_size;
}